// EegFunctionColumnAsyn_14_5m128_28355374088691
// MI455X (gfx1250) — compile-verified
//
#include <hip/hip_runtime.h>

// WTA-LIF 6-layer SNN scan over T=128, B=1024.
// One WG = 16 batch rows, 6 waves, wave l owns layer l+1.
// fp32 WMMA (16x16x4) for exact arithmetic; spikes in LDS; v/t in VGPRs.
// B fragments explicitly double-buffered to overlap L2 loads with WMMA issue.

typedef float v2f __attribute__((ext_vector_type(2)));
typedef float v8f __attribute__((ext_vector_type(8)));

#define PITCH 132                 // floats per LDS spike row (conflict-free for b64 loads)
#define SEGF  (16 * PITCH)        // one spike segment: 16 rows
#define LDS_FLOATS (6 * SEGF)     // s0..s5 -> 50688 bytes

__device__ __forceinline__ v8f wmma4(v2f a, v2f b, v8f c) {
  // D = A(16x4 f32) x B(4x16 f32) + C(16x16 f32)
  return __builtin_amdgcn_wmma_f32_16x16x4_f32(false, a, false, b, (short)0, c,
                                               false, false);
}

// Accumulate S(16xK) @ W(NT*16 x K)^T into acc[0..NT).
// A from LDS, B from global (L2-resident). Software-pipelined: while the 8
// WMMAs of K-step k issue, the b64 loads for K-step k+1 are already in flight.
// K must be a multiple of 8 (64 or 128 here).
template <int NT>
__device__ __forceinline__ void gemm_seg(const float* __restrict__ W, int K,
                                         const float* __restrict__ Aseg,
                                         int lane, v8f* __restrict__ acc) {
  const int cn   = lane & 15;          // A: row M, B: col N within tile
  const int koff = (lane >> 4) << 1;   // lanes 16-31 hold K+2,K+3
  const float* arow = Aseg + cn * PITCH + koff;
  const float* wrow = W + cn * K + koff;

  // prologue: stage k = 0
  v2f a0 = *(const v2f*)(arow);
  v2f b0[NT];
#pragma unroll
  for (int tt = 0; tt < NT; ++tt) b0[tt] = *(const v2f*)(wrow + tt * 16 * K);

  for (int k0 = 0; k0 < K; k0 += 8) {
    const int k1 = k0 + 4;                       // always < K (K % 8 == 0)
    v2f a1 = *(const v2f*)(arow + k1);
    v2f b1[NT];
#pragma unroll
    for (int tt = 0; tt < NT; ++tt)
      b1[tt] = *(const v2f*)(wrow + tt * 16 * K + k1);
#pragma unroll
    for (int tt = 0; tt < NT; ++tt) acc[tt] = wmma4(a0, b0[tt], acc[tt]);

    const int k2 = (k1 + 4 < K) ? (k1 + 4) : 0;  // final iter: dummy wrap load
    a0 = *(const v2f*)(arow + k2);
#pragma unroll
    for (int tt = 0; tt < NT; ++tt)
      b0[tt] = *(const v2f*)(wrow + tt * 16 * K + k2);
#pragma unroll
    for (int tt = 0; tt < NT; ++tt) acc[tt] = wmma4(a1, b1[tt], acc[tt]);
  }
}

// Layer-1 input term: x[:, :, t] (16x14, zero-padded to 16) @ W1b(64x14)^T.
// Fully unrolled; compiler schedules the guarded scalar loads ahead of WMMAs.
__device__ __forceinline__ void gemm_l1_input(const float* __restrict__ x,
                                              const float* __restrict__ W1b,
                                              int b0, int t, int lane,
                                              v8f* __restrict__ acc) {
  const int cn   = lane & 15;
  const int koff = (lane >> 4) << 1;
#pragma unroll
  for (int k0 = 0; k0 < 16; k0 += 4) {
    const int k = k0 + koff;
    v2f a;
    a.x = (k < 14)     ? x[((b0 + cn) * 14 + k) * 128 + t]       : 0.0f;
    a.y = (k + 1 < 14) ? x[((b0 + cn) * 14 + (k + 1)) * 128 + t] : 0.0f;
    v2f b[4];
#pragma unroll
    for (int tt = 0; tt < 4; ++tt) {
      const int n = (tt << 4) + cn;
      b[tt].x = (k < 14)     ? W1b[n * 14 + k]       : 0.0f;
      b[tt].y = (k + 1 < 14) ? W1b[n * 14 + (k + 1)] : 0.0f;
    }
#pragma unroll
    for (int tt = 0; tt < 4; ++tt) acc[tt] = wmma4(a, b[tt], acc[tt]);
  }
}

// v already holds v_new = 0.8*v_old + I (WMMA accumulated into it).
// WTA argmax per row, spike out (masked), hard reset (unmasked), timer update,
// spike written to LDS segment Sout. C-layout: VGPR j = row j (lanes 0-15) /
// row j+8 (lanes 16-31), lane&15 = column within 16-wide tile.
template <int NT, bool HAS_T>
__device__ __forceinline__ void neuron_step(v8f* __restrict__ v,
                                            v8f* __restrict__ tc,
                                            float* __restrict__ Sout, int lane) {
  const int cn = lane & 15;
  const int hi = lane >> 4;
#pragma unroll
  for (int j = 0; j < 8; ++j) {
    float best = -3.4e38f;
    int bestc = 0x7fffffff;
#pragma unroll
    for (int tt = 0; tt < NT; ++tt) {
      const float vv = v[tt][j];
      if (vv > best) { best = vv; bestc = (tt << 4) + cn; }   // strict > keeps lowest col
    }
#pragma unroll
    for (int m = 1; m < 16; m <<= 1) {   // butterfly within each 16-lane half
      const float ov = __shfl_xor(best, m, 32);
      const int   oc = __shfl_xor(bestc, m, 32);
      if (ov > best || (ov == best && oc < bestc)) { best = ov; bestc = oc; }
    }
    const int row = j + (hi << 3);
#pragma unroll
    for (int tt = 0; tt < NT; ++tt) {
      const float vv = v[tt][j];
      const int c = (tt << 4) + cn;
      const float o = (c == bestc && vv >= 1.2f) ? 1.0f : 0.0f;  // spike_out (masked)
      v[tt][j] = (vv >= 1.2f) ? 0.0f : vv;                        // hard reset (unmasked)
      if (HAS_T) tc[tt][j] = (o != 0.0f) ? 0.0f : (tc[tt][j] + 1.0f);
      Sout[row * PITCH + c] = o;
    }
  }
}

__global__ __launch_bounds__(192) void snn_wta_scan(
    const float* __restrict__ x,
    const float* __restrict__ W1b, const float* __restrict__ W1i, const float* __restrict__ R21,
    const float* __restrict__ W2b, const float* __restrict__ W2i, const float* __restrict__ R32,
    const float* __restrict__ W3b, const float* __restrict__ W3i, const float* __restrict__ R43,
    const float* __restrict__ W4b, const float* __restrict__ W4i, const float* __restrict__ R54,
    const float* __restrict__ W5b, const float* __restrict__ W5i, const float* __restrict__ R65,
    const float* __restrict__ W6b, const float* __restrict__ W6i,
    float* __restrict__ out) {
  extern __shared__ __align__(16) float S[];   // s0..s5, pitch 132, 50688 B
  const int tid  = threadIdx.x;
  const int wave = tid >> 5;
  const int lane = tid & 31;
  const int b0   = blockIdx.x << 4;

  // Per-wave layer config: up to 3 (weight, K, spike-segment) inputs.
  const float *WA, *WB, *WC = nullptr;
  int KA, KB, KC = 0, sA, sB, sC = 0, oseg;
  switch (wave) {
    case 0:  WA = W1i; KA = 64;  sA = 0; WB = R21; KB = 128; sB = 1; oseg = 0; break;
    case 1:  WA = W2b; KA = 64;  sA = 0; WB = W2i; KB = 128; sB = 1; WC = R32; KC = 128; sC = 2; oseg = 1; break;
    case 2:  WA = W3b; KA = 128; sA = 1; WB = W3i; KB = 128; sB = 2; WC = R43; KC = 128; sC = 3; oseg = 2; break;
    case 3:  WA = W4b; KA = 128; sA = 2; WB = W4i; KB = 128; sB = 3; WC = R54; KC = 128; sC = 4; oseg = 3; break;
    case 4:  WA = W5b; KA = 128; sA = 3; WB = W5i; KB = 128; sB = 4; WC = R65; KC = 128; sC = 5; oseg = 4; break;
    default: WA = W6b; KA = 128; sA = 4; WB = W6i; KB = 128; sB = 5; oseg = 5; break;
  }

  // Warm weights toward L2 (global_prefetch_b8 on gfx1250).
  __builtin_prefetch(WA + (lane << 6), 0, 3);
  __builtin_prefetch(WB + (lane << 6), 0, 3);
  if (WC) __builtin_prefetch(WC + (lane << 6), 0, 3);

  for (int i = tid; i < LDS_FLOATS; i += 192) S[i] = 0.0f;  // s(t=-1) = 0

  v8f v[8], tc[8];
  const v8f z = {0, 0, 0, 0, 0, 0, 0, 0};
#pragma unroll
  for (int i = 0; i < 8; ++i) { v[i] = z; tc[i] = z; }

  __syncthreads();

  const float* SA = S + sA * SEGF;
  const float* SB = S + sB * SEGF;
  const float* SC = S + sC * SEGF;
  float* SO = S + oseg * SEGF;

  for (int t = 0; t < 128; ++t) {
    // leak: v -= v/TAU  ==  v *= 0.8; WMMA accumulates I directly into v.
#pragma unroll
    for (int i = 0; i < 8; ++i) v[i] = v[i] * 0.8f;

    if (wave == 0) {
      gemm_l1_input(x, W1b, b0, t, lane, v);
      gemm_seg<4>(WA, KA, SA, lane, v);
      gemm_seg<4>(WB, KB, SB, lane, v);
    } else {
      gemm_seg<8>(WA, KA, SA, lane, v);
      gemm_seg<8>(WB, KB, SB, lane, v);
      if (WC) gemm_seg<8>(WC, KC, SC, lane, v);
    }

    __syncthreads();  // all reads of s(t-1) complete

    if (wave == 0) neuron_step<4, false>(v, tc, SO, lane);
    else           neuron_step<8, true >(v, tc, SO, lane);

    __syncthreads();  // all writes of s(t) complete
  }

  // Outputs: exp(-t/100) for layers 2..6 (waves 1..5), each (1024,128).
  if (wave >= 1) {
    const int li = wave - 1;
    const int cn = lane & 15;
    const int hi = lane >> 4;
#pragma unroll
    for (int tt = 0; tt < 8; ++tt)
#pragma unroll
      for (int j = 0; j < 8; ++j) {
        const int row = j + (hi << 3);
        out[li * (1024 * 128) + (b0 + row) * 128 + (tt * 16 + cn)] =
            expf(-tc[tt][j] / 100.0f);
      }
  }
}

extern "C" void kernel_launch(void* const* d_in, const int* in_sizes, int n_in,
                              void* d_out, int out_size, void* d_ws, size_t ws_size,
                              hipStream_t stream) {
  (void)in_sizes; (void)n_in; (void)out_size; (void)d_ws; (void)ws_size;
  const float* x   = (const float*)d_in[0];
  const float* W1b = (const float*)d_in[1];
  const float* W1i = (const float*)d_in[2];
  const float* R21 = (const float*)d_in[3];
  const float* W2b = (const float*)d_in[4];
  const float* W2i = (const float*)d_in[5];
  const float* R32 = (const float*)d_in[6];
  const float* W3b = (const float*)d_in[7];
  const float* W3i = (const float*)d_in[8];
  const float* R43 = (const float*)d_in[9];
  const float* W4b = (const float*)d_in[10];
  const float* W4i = (const float*)d_in[11];
  const float* R54 = (const float*)d_in[12];
  const float* W5b = (const float*)d_in[13];
  const float* W5i = (const float*)d_in[14];
  const float* R65 = (const float*)d_in[15];
  const float* W6b = (const float*)d_in[16];
  const float* W6i = (const float*)d_in[17];
  float* out = (float*)d_out;

  const size_t smem = (size_t)LDS_FLOATS * sizeof(float);  // 50688 B < 64 KB
  snn_wta_scan<<<1024 / 16, 192, smem, stream>>>(
      x, W1b, W1i, R21, W2b, W2i, R32, W3b, W3i, R43,
      W4b, W4i, R54, W5b, W5i, R65, W6b, W6i, out);
}